// penalty_focal_loss2_1211180778083
// MI455X (gfx1250) — compile-verified
//
#include <hip/hip_runtime.h>
#include <stdint.h>

// ---------------------------------------------------------------------------
// Penalty focal loss (gamma=2), B=2^21 rows x C=32 classes, fp32 -> scalar.
// Memory-bound streaming kernel: 512MB in / 4B out => ~22us floor @ 23.3TB/s.
// CDNA5 path used: GLOBAL_LOAD_ASYNC_TO_LDS_B128 (ASYNCcnt double buffering),
// wave32 lane-per-row processing, 2-phase-conflict-free DS_LOAD_B128 layout.
// WMMA deliberately not used: no matmul structure; 16x redundant MACs would
// only burn power on a bandwidth-bound op.
// ---------------------------------------------------------------------------

#define PFL_NBLK        1024
#define WAVES_PER_BLOCK 8
#define THREADS         (WAVES_PER_BLOCK * 32)
#define TILE_ROWS       32
#define ROW_BYTES       128                          // 32 classes * 4B
#define TILE_BYTES      (TILE_ROWS * ROW_BYTES)      // 4096 B per array per tile
#define WAVE_LDS        (2 * 2 * TILE_BYTES)         // dbl-buffer * (P+T) = 16 KB
#define PM_LDS          4096                         // 32x32 f32 penalty matrix
#define SMEM_BYTES      (PM_LDS + WAVES_PER_BLOCK * WAVE_LDS)  // 135168 B

__device__ __forceinline__ float fast_exp2(float x) {
#if __has_builtin(__builtin_amdgcn_exp2f)
  return __builtin_amdgcn_exp2f(x);          // v_exp_f32
#else
  return exp2f(x);
#endif
}
__device__ __forceinline__ float fast_log2(float x) {
#if __has_builtin(__builtin_amdgcn_logf)
  return __builtin_amdgcn_logf(x);           // v_log_f32
#else
  return log2f(x);
#endif
}
__device__ __forceinline__ float fast_rcp(float x) {
#if __has_builtin(__builtin_amdgcn_rcpf)
  return __builtin_amdgcn_rcpf(x);           // v_rcp_f32
#else
  return 1.0f / x;
#endif
}

// Copy one 4KB tile (32 rows x 128B) global -> LDS with the async data mover.
// Global side is perfectly linear: granule n = i*32+lane, 512B per instruction.
// LDS side is chunk-transposed: 16B chunk c of row r goes to 16B slot c*32 + r, so that the
// consumer (lane = row) reads ds_load_b128 at start-bank 4*lane: exactly
// 2-phase bank-conflict-free for wave32.
__device__ __forceinline__ void issue_tile_async(const float* gbase, unsigned gOff,
                                                 unsigned ldsBase, unsigned lane) {
#pragma unroll
  for (unsigned i = 0; i < 8; ++i) {
    unsigned n   = i * 32u + lane;
    unsigned vof = gOff + n * 16u;
    unsigned lds = ldsBase + ((((n & 7u) << 5) | (n >> 3)) << 4);
    asm volatile("global_load_async_to_lds_b128 %0, %1, %2"
                 :
                 : "v"(lds), "v"(vof), "s"(gbase)
                 : "memory");
  }
}

__global__ void __launch_bounds__(THREADS)
pfl_partials(const float* __restrict__ predict,
             const float* __restrict__ target,
             const float* __restrict__ pm,
             float* __restrict__ partials,
             int tilesPerWave)
{
  extern __shared__ __align__(16) char smem[];
  // Low 32 bits of a generic LDS address are the wave-relative LDS byte
  // address (CDNA5 aperture layout) -> usable directly by async-to-LDS ops.
  const unsigned smemBase = (unsigned)(uintptr_t)smem;
  float* sPM = (float*)smem;

  const unsigned lane = threadIdx.x & 31u;
  const unsigned wave = threadIdx.x >> 5;

  // Stage 32x32 penalty matrix once per block.
  for (int i = (int)threadIdx.x; i < 1024; i += THREADS) sPM[i] = pm[i];
  __syncthreads();

  const unsigned waveLds = smemBase + PM_LDS + wave * WAVE_LDS;
  const unsigned gWave   = blockIdx.x * WAVES_PER_BLOCK + wave;     // 0..8191
  // Each wave owns tilesPerWave contiguous 32-row tiles (256MB max => u32 ok).
  const unsigned base = gWave * (unsigned)(tilesPerWave * TILE_BYTES);

  // Prefetch tile 0 into buffer 0 (predict + target: 16 async ops).
  issue_tile_async(predict, base, waveLds,              lane);
  issue_tile_async(target,  base, waveLds + TILE_BYTES, lane);

  const float L2E = 1.4426950408889634f;   // log2(e)
  const float LN2 = 0.6931471805599453f;   // ln(2)
  float acc = 0.0f;

#pragma unroll 1
  for (int t = 0; t < tilesPerWave; ++t) {
    const unsigned buf = waveLds + (unsigned)(t & 1) * (2u * TILE_BYTES);
    if (t + 1 < tilesPerWave) {
      // Prefetch next tile into the other buffer, then wait until only those
      // 16 new async ops are outstanding -> current tile has landed in LDS.
      const unsigned nbuf = waveLds + (unsigned)((t + 1) & 1) * (2u * TILE_BYTES);
      const unsigned nOff = base + (unsigned)(t + 1) * TILE_BYTES;
      issue_tile_async(predict, nOff, nbuf,              lane);
      issue_tile_async(target,  nOff, nbuf + TILE_BYTES, lane);
      asm volatile("s_wait_asynccnt 16" ::: "memory");
    } else {
      asm volatile("s_wait_asynccnt 0" ::: "memory");
    }

    // Lane `lane` owns one full row of this tile.
    const char* smemG = smem + (buf - smemBase);
    const float4* __restrict__ P4 = (const float4*)(smemG);
    const float4* __restrict__ T4 = (const float4*)(smemG + TILE_BYTES);

    float p[32];
    float ptrue = 0.0f;
    int   tid   = 0;
#pragma unroll
    for (int c = 0; c < 8; ++c) {
      float4 v = P4[c * 32 + lane];        // ds_load_b128, 2-phase conflict-free
      float4 u = T4[c * 32 + lane];
      p[4 * c + 0] = v.x; p[4 * c + 1] = v.y;
      p[4 * c + 2] = v.z; p[4 * c + 3] = v.w;
      bool o;
      o = u.x > 0.5f; ptrue = o ? v.x : ptrue; tid = o ? 4 * c + 0 : tid;
      o = u.y > 0.5f; ptrue = o ? v.y : ptrue; tid = o ? 4 * c + 1 : tid;
      o = u.z > 0.5f; ptrue = o ? v.z : ptrue; tid = o ? 4 * c + 2 : tid;
      o = u.w > 0.5f; ptrue = o ? v.w : ptrue; tid = o ? 4 * c + 3 : tid;
    }

    // argmax(predict): strict '>' scan keeps first index on ties (jnp.argmax).
    float best = p[0]; int pid = 0;
#pragma unroll
    for (int i = 1; i < 32; ++i) {
      bool g = p[i] > best;
      best = g ? p[i] : best;
      pid  = g ? i : pid;
    }

    // Stable softmax denominator: sum exp(p - max) = sum exp2(p*L2E - max*L2E).
    float mh = best * L2E;
    float s  = 0.0f;
#pragma unroll
    for (int i = 0; i < 32; ++i) s += fast_exp2(fmaf(p[i], L2E, -mh));

    float d     = ptrue - best;                  // ln(e_true)
    float ln_s  = fast_log2(s) * LN2;
    float log_p = d - ln_s;                      // ln(prob), <= 0
    float prob  = fast_exp2(d * L2E) * fast_rcp(s);
    float omp   = 1.0f - prob;
    float alpha = sPM[(tid << 5) + pid];         // LDS gather, L2-free
    acc -= alpha * omp * omp * log_p;            // loss >= 0
  }

  // Deterministic wave32 butterfly reduction; one partial per wave.
#pragma unroll
  for (int off = 16; off > 0; off >>= 1) acc += __shfl_xor(acc, off, 32);
  if (lane == 0) partials[gWave] = acc;
}

// Deterministic final reduction: fixed-order strided sums + shared-mem tree.
__global__ void __launch_bounds__(256)
pfl_reduce(const float* __restrict__ partials, float* __restrict__ out,
           int n, float invB)
{
  __shared__ float sm[256];
  float a = 0.0f;
  for (int i = (int)threadIdx.x; i < n; i += 256) a += partials[i];
  sm[threadIdx.x] = a;
  __syncthreads();
  for (int s = 128; s > 0; s >>= 1) {
    if ((int)threadIdx.x < s) sm[threadIdx.x] += sm[threadIdx.x + s];
    __syncthreads();
  }
  if (threadIdx.x == 0) out[0] = sm[0] * invB;
}

extern "C" void kernel_launch(void* const* d_in, const int* in_sizes, int n_in,
                              void* d_out, int out_size, void* d_ws, size_t ws_size,
                              hipStream_t stream) {
  const float* predict  = (const float*)d_in[0];
  const float* target   = (const float*)d_in[1];
  const float* pm       = (const float*)d_in[2];
  float*       out      = (float*)d_out;
  float*       partials = (float*)d_ws;   // 8192 floats = 32 KB scratch

  const long long BC    = (long long)in_sizes[0];   // B*C
  const int Brows       = (int)(BC / 32);
  const int nWaves      = PFL_NBLK * WAVES_PER_BLOCK;          // 8192
  const int tilesPerWave = Brows / (nWaves * TILE_ROWS);       // 8 for B=2^21

  pfl_partials<<<PFL_NBLK, THREADS, SMEM_BYTES, stream>>>(
      predict, target, pm, partials, tilesPerWave);
  pfl_reduce<<<1, 256, 0, stream>>>(partials, out, nWaves, 1.0f / (float)Brows);
}